// LowRankINRLayer_36782099923162
// MI455X (gfx1250) — compile-verified
//
#include <hip/hip_runtime.h>

// Problem constants (match reference)
#define B_  16
#define N_  8192
#define DI_ 512
#define DO_ 512
#define R_  32

typedef __attribute__((ext_vector_type(2))) float v2f;
typedef __attribute__((ext_vector_type(8))) float v8f;

// Fused low-rank layer:  out = relu( (x @ v^T) @ W^T )
// One workgroup = 8 waves, handles one batch b and 128 rows of n.
// Each wave computes 16 rows end-to-end, staging the 16x32 xv tile in LDS.
__global__ __launch_bounds__(256) void lowrank_inr_kernel(
    const float* __restrict__ x,   // [B, N, DI]
    const float* __restrict__ v,   // [B, R, DI]
    const float* __restrict__ W,   // [DO, R]
    float* __restrict__ out)       // [B, N, DO]
{
    // xv tile per wave: 16 rows x 32 (R) cols, pitch 34 floats (even pitch ->
    // 8B-aligned v2f reads; 34 mod 64-bank layout is conflict-free).
    __shared__ float xv[8][16][34];

    const int tid  = threadIdx.x;
    const int wave = tid >> 5;
    const int lane = tid & 31;
    const int l16  = lane & 15;
    const int hi   = lane >> 4;        // 0: lanes 0-15, 1: lanes 16-31

    const int b    = blockIdx.y;
    const int row0 = blockIdx.x * 128 + wave * 16;   // this wave's 16 rows

    // ---------------- Stage 1: xv[16 x 32] = x_tile[16 x 512] @ v[b]^T ----------------
    // A fragment (16x4 f32): lane l16 holds row m=l16, k = 2*hi + {0,1} (consecutive).
    // B fragment (4x16 f32): lane l16 holds col n=l16 (= r), k = 2*hi + {0,1}.
    const float* xrow = x + ((size_t)b * N_ + row0 + l16) * DI_ + 2 * hi;
    const float* vrow = v + ((size_t)b * R_ + l16) * DI_ + 2 * hi;   // r tile 0: r=l16
    const float* vrow1 = vrow + (size_t)16 * DI_;                    // r tile 1: r=16+l16

    v8f c0a = {}, c0b = {}, c1a = {}, c1b = {};   // 4 independent WMMA chains
    #pragma unroll 4
    for (int k = 0; k < DI_; k += 8) {
        // stream-prefetch x ~8 k-steps ahead
        __builtin_prefetch(xrow + k + 64, 0, 1);
        v2f aA  = *(const v2f*)(xrow  + k);
        v2f aB  = *(const v2f*)(xrow  + k + 4);
        v2f b0A = *(const v2f*)(vrow  + k);
        v2f b0B = *(const v2f*)(vrow  + k + 4);
        v2f b1A = *(const v2f*)(vrow1 + k);
        v2f b1B = *(const v2f*)(vrow1 + k + 4);
        c0a = __builtin_amdgcn_wmma_f32_16x16x4_f32(false, aA, false, b0A, (short)0, c0a, false, false);
        c1a = __builtin_amdgcn_wmma_f32_16x16x4_f32(false, aA, false, b1A, (short)0, c1a, false, false);
        c0b = __builtin_amdgcn_wmma_f32_16x16x4_f32(false, aB, false, b0B, (short)0, c0b, false, false);
        c1b = __builtin_amdgcn_wmma_f32_16x16x4_f32(false, aB, false, b1B, (short)0, c1b, false, false);
    }
    v8f c0 = c0a + c0b;
    v8f c1 = c1a + c1b;

    // Spill xv tile to LDS. C/D layout: VGPR j, lane -> row m = j + 8*hi, col = l16.
    #pragma unroll
    for (int j = 0; j < 8; ++j) {
        xv[wave][j + 8 * hi][l16]      = c0[j];
        xv[wave][j + 8 * hi][16 + l16] = c1[j];
    }
    __syncthreads();

    // ---------------- Stage 2: out_tile[16 x 512] = relu(xv[16 x 32] @ W^T) ----------------
    // Hoist all 8 A fragments (K=32) from LDS once; reuse across 32 column tiles.
    v2f a2[8];
    #pragma unroll
    for (int kk = 0; kk < 8; ++kk) {
        a2[kk] = *(const v2f*)(&xv[wave][l16][4 * kk + 2 * hi]);
    }

    const size_t outbase = ((size_t)b * N_ + row0 + 8 * hi) * DO_ + l16;

    #pragma unroll 2
    for (int oc = 0; oc < 32; oc += 2) {
        // B fragments: lane l16 holds col o = 16*oc + l16, k = 4*kk + 2*hi + {0,1};
        // W is [DO, R] row-major so those two k-values are contiguous.
        const float* wp0 = W + ((size_t)(16 * (oc + 0) + l16)) * R_ + 2 * hi;
        const float* wp1 = W + ((size_t)(16 * (oc + 1) + l16)) * R_ + 2 * hi;
        v8f acc0 = {}, acc1 = {};
        #pragma unroll
        for (int kk = 0; kk < 8; ++kk) {
            v2f bw0 = *(const v2f*)(wp0 + 4 * kk);
            v2f bw1 = *(const v2f*)(wp1 + 4 * kk);
            acc0 = __builtin_amdgcn_wmma_f32_16x16x4_f32(false, a2[kk], false, bw0, (short)0, acc0, false, false);
            acc1 = __builtin_amdgcn_wmma_f32_16x16x4_f32(false, a2[kk], false, bw1, (short)0, acc1, false, false);
        }
        // ReLU + store. Row m = j + 8*hi, col o = 16*oc + l16.
        float* op0 = out + outbase + 16 * (oc + 0);
        float* op1 = out + outbase + 16 * (oc + 1);
        #pragma unroll
        for (int j = 0; j < 8; ++j) {
            float r0 = acc0[j] > 0.f ? acc0[j] : 0.f;
            float r1 = acc1[j] > 0.f ? acc1[j] : 0.f;
            op0[(size_t)j * DO_] = r0;
            op1[(size_t)j * DO_] = r1;
        }
    }
}

extern "C" void kernel_launch(void* const* d_in, const int* in_sizes, int n_in,
                              void* d_out, int out_size, void* d_ws, size_t ws_size,
                              hipStream_t stream) {
    const float* x = (const float*)d_in[0];   // [B, N, DI]
    const float* v = (const float*)d_in[1];   // [B, R, DI]
    const float* W = (const float*)d_in[2];   // [DO, R]
    float* out     = (float*)d_out;           // [B, N, DO]
    (void)in_sizes; (void)n_in; (void)out_size; (void)d_ws; (void)ws_size;

    dim3 grid(N_ / 128, B_);   // 64 x 16 workgroups
    dim3 block(256);           // 8 waves (wave32)
    lowrank_inr_kernel<<<grid, block, 0, stream>>>(x, v, W, out);
}